// Patches_19533511262778
// MI455X (gfx1250) — compile-verified
//
#include <hip/hip_runtime.h>
#include <hip/hip_bf16.h>
#include <stdint.h>

// -----------------------------------------------------------------------------
// Patch extraction [128,256,256,3] f32 -> [128,1024,192] f32.
// Pure permutation, 0 FLOPs, ~201 MB traffic -> ~8.6 us roofline @ 23.3 TB/s.
//
// Tile = (b, gh): contiguous 24KB in input AND contiguous 24KB in output; the
// shuffle is interior only (96B chunks move intact). Routed through LDS with
// CDNA5 async global<->LDS ops (ASYNCcnt path): linear global reads with
// permuted LDS destinations, then linear LDS -> linear global writes. No VGPR
// data staging; both global streams are perfectly coalesced B128.
//
// Geometry: 192 threads (6 waves), 8 iterations of 192 float4s. With
// i = k*192 + tid, the patch-row pr == k (loop constant) and the only integer
// division left is tid/6 once per thread.
// -----------------------------------------------------------------------------

#define THREADS 192
#define TILE_F4 1536   // float4s per 24KB tile (6144 floats)
#define ITERS   (TILE_F4 / THREADS)  // 8

__device__ __forceinline__ void async_load_b128(uint32_t lds_byte_addr,
                                                const void* gaddr) {
  // global_load_async_to_lds_b128  vdst(lds addr), vaddr(64b), off
  asm volatile("global_load_async_to_lds_b128 %0, %1, off"
               :
               : "v"(lds_byte_addr), "v"(gaddr)
               : "memory");
}

__device__ __forceinline__ void async_store_b128(const void* gaddr,
                                                 uint32_t lds_byte_addr) {
  // global_store_async_from_lds_b128  vaddr(64b), vsrc(lds addr), off
  asm volatile("global_store_async_from_lds_b128 %0, %1, off"
               :
               : "v"(gaddr), "v"(lds_byte_addr)
               : "memory");
}

__device__ __forceinline__ void wait_asynccnt0() {
#if __has_builtin(__builtin_amdgcn_s_wait_asynccnt)
  __builtin_amdgcn_s_wait_asynccnt(0);
#else
  asm volatile("s_wait_asynccnt 0x0" ::: "memory");
#endif
}

__global__ __launch_bounds__(THREADS) void Patches_19533511262778_kernel(
    const float* __restrict__ in, float* __restrict__ out) {
  __shared__ float4 tile[TILE_F4];  // 24 KB

  // Raw LDS byte address (low 32 bits of the generic pointer are the
  // wave-relative LDS offset on gfx1250).
  const uint32_t lds_base = (uint32_t)(uintptr_t)&tile[0];

  const int t = blockIdx.x;  // tile id = b*32 + gh, 4096 tiles total
  const float4* __restrict__ src = (const float4*)in + (size_t)t * TILE_F4;
  float4* __restrict__ dst       = (float4*)out + (size_t)t * TILE_F4;

  const int tid = threadIdx.x;          // 0..191 = float4 within an image row
  const int gw  = tid / 6;              // patch column (0..31)
  const int q   = tid - 6 * gw;         // float4 within the 24-float chunk
  // output-local float4 for (pr=k, gw, q): o = gw*48 + k*6 + q
  const uint32_t lds_o0 = lds_base + (uint32_t)(gw * 48 + q) * 16u;

  // ---- Phase 1: contiguous global -> permuted LDS (async, no VGPR data) ----
#pragma unroll
  for (int k = 0; k < ITERS; ++k) {
    async_load_b128(lds_o0 + (uint32_t)(k * 6) * 16u, src + k * THREADS + tid);
  }
  wait_asynccnt0();     // this wave's LDS deposits are complete
  __syncthreads();      // all waves' deposits visible block-wide

  // ---- Phase 2: contiguous LDS -> contiguous global (async) ----
#pragma unroll
  for (int k = 0; k < ITERS; ++k) {
    const int i = k * THREADS + tid;
    async_store_b128(dst + i, lds_base + (uint32_t)i * 16u);
  }
  wait_asynccnt0();     // drain before wave exit (S_ENDPGM also wait-idles)
}

extern "C" void kernel_launch(void* const* d_in, const int* in_sizes, int n_in,
                              void* d_out, int out_size, void* d_ws, size_t ws_size,
                              hipStream_t stream) {
  (void)in_sizes; (void)n_in; (void)d_ws; (void)ws_size; (void)out_size;
  const float* images = (const float*)d_in[0];
  float* out = (float*)d_out;

  // 128 batches * 32 patch-row groups = 4096 tiles, one block each.
  dim3 grid(4096), block(THREADS);
  Patches_19533511262778_kernel<<<grid, block, 0, stream>>>(images, out);
}